// GCN_59313498358227
// MI455X (gfx1250) — compile-verified
//
#include <hip/hip_runtime.h>
#include <hip/hip_bf16.h>
#include <math.h>

typedef float v2f __attribute__((ext_vector_type(2)));
typedef float v8f __attribute__((ext_vector_type(8)));

// ---------------------------------------------------------------------------
// Zero fill (grid-stride)
// ---------------------------------------------------------------------------
__global__ void zero_kernel(float* __restrict__ p, long long n) {
    long long i = (long long)blockIdx.x * blockDim.x + threadIdx.x;
    long long stride = (long long)gridDim.x * blockDim.x;
    for (; i < n; i += stride) p[i] = 0.0f;
}

// ---------------------------------------------------------------------------
// Degree accumulation: deg_out[src[e]] += 1, deg_in[dst[e]] += 1
// ---------------------------------------------------------------------------
__global__ void degree_kernel(const int* __restrict__ src, const int* __restrict__ dst,
                              float* __restrict__ deg_out, float* __restrict__ deg_in, int E) {
    int e = blockIdx.x * blockDim.x + threadIdx.x;
    if (e >= E) return;
    atomicAdd(&deg_out[src[e]], 1.0f);
    atomicAdd(&deg_in[dst[e]], 1.0f);
}

// In-place: deg -> clip(deg,1)^-1/2
__global__ void norm_kernel(float* __restrict__ a, float* __restrict__ b, int N) {
    int i = blockIdx.x * blockDim.x + threadIdx.x;
    if (i >= N) return;
    a[i] = rsqrtf(fmaxf(a[i], 1.0f));
    b[i] = rsqrtf(fmaxf(b[i], 1.0f));
}

// ---------------------------------------------------------------------------
// WMMA GEMM: Y[m, :] = (X[m, :] * nsrc[m]) @ W     (X: [M,KDIM], W: [KDIM,NOUT])
// One wave32 computes a 16-row stripe across all NOUT columns using
// V_WMMA_F32_16X16X4_F32 (fp32 in/out, exact-precision match to reference).
//
// A fragment (16x4 f32): lanes 0-15 hold {K=k,k+1} of row m=lane,
//                        lanes 16-31 hold {K=k+2,k+3} of row m=lane-16.
// B fragment (4x16 f32): lanes 0-15 hold {B[k][n],B[k+1][n]}, n=lane,
//                        lanes 16-31 hold {B[k+2][n],B[k+3][n]}, n=lane-16.
// C/D (16x16 f32): VGPR r -> row mBase + r + 8*(lane>>4), col = lane&15.
// ---------------------------------------------------------------------------
template <int KDIM, int NOUT>
__global__ void gemm_wmma_kernel(const float* __restrict__ X, const float* __restrict__ W,
                                 const float* __restrict__ nsrc, float* __restrict__ Y, int M) {
    constexpr int NT = (NOUT + 15) / 16;
    const int gtid = blockIdx.x * blockDim.x + threadIdx.x;
    const int wave = gtid >> 5;
    const int lane = threadIdx.x & 31;
    const int mBase = wave * 16;
    if (mBase >= M) return;  // wave-uniform exit; EXEC stays all-1s inside

    const int half = lane >> 4;   // 0: K pair (k,k+1), 1: (k+2,k+3)
    const int l15  = lane & 15;
    const int m    = mBase + l15;
    const float s  = nsrc[m];
    const float* __restrict__ xrow = X + (size_t)m * KDIM;

    v8f acc[NT];
#pragma unroll
    for (int t = 0; t < NT; ++t)
#pragma unroll
        for (int r = 0; r < 8; ++r) acc[t][r] = 0.0f;

    for (int k = 0; k < KDIM; k += 4) {
        v2f a;
        a.x = xrow[k + 2 * half + 0] * s;
        a.y = xrow[k + 2 * half + 1] * s;
#pragma unroll
        for (int t = 0; t < NT; ++t) {
            const int n  = t * 16 + l15;
            const int nn = (n < NOUT) ? n : 0;   // value-select, no divergence
            v2f b;
            float b0 = W[(size_t)(k + 2 * half + 0) * NOUT + nn];
            float b1 = W[(size_t)(k + 2 * half + 1) * NOUT + nn];
            b.x = (n < NOUT) ? b0 : 0.0f;
            b.y = (n < NOUT) ? b1 : 0.0f;
            acc[t] = __builtin_amdgcn_wmma_f32_16x16x4_f32(
                /*neg_a=*/false, a, /*neg_b=*/false, b,
                /*c_mod=*/(short)0, acc[t], /*reuse_a=*/false, /*reuse_b=*/false);
        }
    }

#pragma unroll
    for (int t = 0; t < NT; ++t) {
#pragma unroll
        for (int r = 0; r < 8; ++r) {
            const int row = mBase + r + 8 * half;
            const int col = t * 16 + l15;
            if (col < NOUT && row < M)
                Y[(size_t)row * NOUT + col] = acc[t][r];
        }
    }
}

// ---------------------------------------------------------------------------
// Edge scatter-add: agg[dst[e], :] += X[src[e], :]
// Each thread owns VEC consecutive floats of one edge; atomics land in L2
// (entire agg buffer << 192MB L2) as non-returning global_atomic_add_f32.
// ---------------------------------------------------------------------------
template <int F, int VEC>
__global__ void scatter_kernel(const float* __restrict__ X, const int* __restrict__ src,
                               const int* __restrict__ dst, float* __restrict__ agg, int E) {
    constexpr int CHUNKS = F / VEC;
    long long tid = (long long)blockIdx.x * blockDim.x + threadIdx.x;
    long long total = (long long)E * CHUNKS;
    if (tid >= total) return;
    const int e = (int)(tid / CHUNKS);
    const int c = (int)(tid % CHUNKS);
    const int sI = src[e];
    const int dI = dst[e];
    const float* __restrict__ xp = X + (size_t)sI * F + c * VEC;
    float* __restrict__ ap = agg + (size_t)dI * F + c * VEC;
#pragma unroll
    for (int j = 0; j < VEC; ++j) atomicAdd(ap + j, xp[j]);
}

// ---------------------------------------------------------------------------
// h = relu(h * ndst[row] + bias[col])   (in place)
// ---------------------------------------------------------------------------
template <int F>
__global__ void bias_relu_kernel(float* __restrict__ h, const float* __restrict__ ndst,
                                 const float* __restrict__ bias, int N) {
    long long tid = (long long)blockIdx.x * blockDim.x + threadIdx.x;
    if (tid >= (long long)N * F) return;
    const int row = (int)(tid / F);
    const int col = (int)(tid % F);
    float v = h[tid] * ndst[row] + bias[col];
    h[tid] = v > 0.0f ? v : 0.0f;
}

// ---------------------------------------------------------------------------
// out[i,:] = log_softmax(agg[i,:] * ndst[i] + bias), row width 40
// ---------------------------------------------------------------------------
__global__ void final_lsm_kernel(const float* __restrict__ agg, const float* __restrict__ ndst,
                                 const float* __restrict__ bias, float* __restrict__ out, int N) {
    int i = blockIdx.x * blockDim.x + threadIdx.x;
    if (i >= N) return;
    const int F = 40;
    const float nd = ndst[i];
    float v[F];
    float mx = -3.4e38f;
#pragma unroll
    for (int f = 0; f < F; ++f) {
        v[f] = agg[(size_t)i * F + f] * nd + bias[f];
        mx = fmaxf(mx, v[f]);
    }
    float sum = 0.0f;
#pragma unroll
    for (int f = 0; f < F; ++f) sum += expf(v[f] - mx);
    const float lse = logf(sum);
#pragma unroll
    for (int f = 0; f < F; ++f) out[(size_t)i * F + f] = v[f] - mx - lse;
}

// ---------------------------------------------------------------------------
// Driver
// ---------------------------------------------------------------------------
extern "C" void kernel_launch(void* const* d_in, const int* in_sizes, int n_in,
                              void* d_out, int out_size, void* d_ws, size_t ws_size,
                              hipStream_t stream) {
    const float* feat = (const float*)d_in[0];
    const int*   src  = (const int*)d_in[1];
    const int*   dst  = (const int*)d_in[2];
    const float* W1   = (const float*)d_in[3];
    const float* b1   = (const float*)d_in[4];
    const float* W2   = (const float*)d_in[5];
    const float* b2   = (const float*)d_in[6];
    const float* W3   = (const float*)d_in[7];
    const float* b3   = (const float*)d_in[8];
    float* out = (float*)d_out;

    const int N = in_sizes[0] / 256;  // 100000
    const int E = in_sizes[1];        // 1600000

    float* ws   = (float*)d_ws;
    float* nsrc = ws;                  // N   (deg_out -> norm_src in place)
    float* ndst = nsrc + N;            // N
    float* bufA = ndst + N;            // N*128  (X1, later X2)
    float* bufB = bufA + (size_t)N * 128;  // N*128  (agg1/y1, agg2/y2)
    float* bufC = bufB + (size_t)N * 128;  // N*40   (X3)
    float* bufD = bufC + (size_t)N * 40;   // N*40   (agg3)

    const int TB = 256;
    auto blocks = [](long long work, int tb) { return (int)((work + tb - 1) / tb); };

    // --- degrees & norms (re-zeroed every call: deterministic) ---
    zero_kernel<<<blocks(2LL * N, TB), TB, 0, stream>>>(nsrc, 2LL * N);
    degree_kernel<<<blocks(E, TB), TB, 0, stream>>>(src, dst, nsrc, ndst, E);
    norm_kernel<<<blocks(N, TB), TB, 0, stream>>>(nsrc, ndst, N);

    const int waves1 = (N + 15) / 16;          // one wave per 16-row stripe
    const int gemmBlocks = (waves1 * 32 + TB - 1) / TB;

    // --- layer 1: (feat*nsrc)@W1 -> scatter -> *ndst + b1, relu ---
    gemm_wmma_kernel<256, 128><<<gemmBlocks, TB, 0, stream>>>(feat, W1, nsrc, bufA, N);
    zero_kernel<<<blocks((long long)N * 128, TB), TB, 0, stream>>>(bufB, (long long)N * 128);
    scatter_kernel<128, 4><<<blocks((long long)E * 32, TB), TB, 0, stream>>>(bufA, src, dst, bufB, E);
    bias_relu_kernel<128><<<blocks((long long)N * 128, TB), TB, 0, stream>>>(bufB, ndst, b1, N);

    // --- layer 2 ---
    gemm_wmma_kernel<128, 128><<<gemmBlocks, TB, 0, stream>>>(bufB, W2, nsrc, bufA, N);
    zero_kernel<<<blocks((long long)N * 128, TB), TB, 0, stream>>>(bufB, (long long)N * 128);
    scatter_kernel<128, 4><<<blocks((long long)E * 32, TB), TB, 0, stream>>>(bufA, src, dst, bufB, E);
    bias_relu_kernel<128><<<blocks((long long)N * 128, TB), TB, 0, stream>>>(bufB, ndst, b2, N);

    // --- layer 3 + log_softmax ---
    gemm_wmma_kernel<128, 40><<<gemmBlocks, TB, 0, stream>>>(bufB, W3, nsrc, bufC, N);
    zero_kernel<<<blocks((long long)N * 40, TB), TB, 0, stream>>>(bufD, (long long)N * 40);
    scatter_kernel<40, 4><<<blocks((long long)E * 10, TB), TB, 0, stream>>>(bufC, src, dst, bufD, E);
    final_lsm_kernel<<<blocks(N, TB), TB, 0, stream>>>(bufD, ndst, b3, out, N);
}